// ModelTree2_1_12515534700682
// MI455X (gfx1250) — compile-verified
//
#include <hip/hip_runtime.h>
#include <hip/hip_bf16.h>
#include <math.h>

// ---------------------------------------------------------------------------
// Problem constants (from the reference)
// ---------------------------------------------------------------------------
#define NN      60000      // graph nodes
#define DD      128        // dynamic size
#define EE      600000     // edges
#define BB      256        // batch
#define FEATSZ  128
#define IN_RNN  257        // D + 1 + FEAT
#define ALPHA_F 0.1f
#define ONE_M_ALPHA 0.9f
#define EPS_F   1e-12f

typedef float v2f __attribute__((ext_vector_type(2)));
typedef float v8f __attribute__((ext_vector_type(8)));

// LDS padding strides (in floats), chosen for conflict-free wave32 reads:
//  - A reads: lane l (half h) hits bank (5*l + K + 2h) % 64 -> no two lanes collide
//  - W reads: half-waves are 16 banks apart (272 % 64 == 16), col-tile base is
//    a multiple of 16 -> the two halves use disjoint bank sets
#define A_STRIDE 133
#define W_STRIDE 136

// ---------------------------------------------------------------------------
// support = act(X) @ W   with 16x16x4 fp32 WMMA (exact fp32 path)
// One block: 16 rows of X, all 128 cols.  8 waves = 8 column tiles.
// ---------------------------------------------------------------------------
__global__ __launch_bounds__(256) void gcn_gemm_wmma(
    const float* __restrict__ X, const float* __restrict__ W,
    float* __restrict__ S, int apply_relu)
{
    __shared__ float sW[128 * W_STRIDE];   // 69,632 B
    __shared__ float sA[16  * A_STRIDE];   //  8,512 B

    const int r0  = blockIdx.x * 16;
    const int tid = threadIdx.x;

    // Stage W[k][n] (row-major 128x128) into padded LDS, coalesced.
    for (int i = tid; i < 128 * 128; i += 256) {
        const int k = i >> 7, n = i & 127;
        sW[k * W_STRIDE + n] = W[i];
    }
    // Stage the 16x128 A tile (optionally fusing the previous layer's ReLU).
    for (int i = tid; i < 16 * 128; i += 256) {
        const int m = i >> 7, k = i & 127;
        float v = X[(size_t)(r0 + m) * DD + k];
        if (apply_relu) v = fmaxf(v, 0.0f);
        sA[m * A_STRIDE + k] = v;
    }
    __syncthreads();

    const int wave = tid >> 5;        // 0..7 -> 16-col tile
    const int lane = tid & 31;
    const int half = lane >> 4;       // 0: lanes 0-15, 1: lanes 16-31
    const int l    = lane & 15;
    const int c0   = wave * 16;

    v8f acc = {0.f, 0.f, 0.f, 0.f, 0.f, 0.f, 0.f, 0.f};

    // K loop: V_WMMA_F32_16X16X4_F32, 32 steps of K=4.
    // A frag (ISA 7.12.2, 32-bit A 16x4): lanes 0-15 hold K=k+0,k+1 in the two
    // VGPRs; lanes 16-31 hold K=k+2,k+3.  B frag mirrored (row of B striped
    // across lanes; halves carry K and K+2).
    for (int k = 0; k < DD; k += 4) {
        const int ka = k + half * 2;
        v2f a, b;
        a.x = sA[l * A_STRIDE + ka];
        a.y = sA[l * A_STRIDE + ka + 1];
        b.x = sW[(ka)     * W_STRIDE + c0 + l];
        b.y = sW[(ka + 1) * W_STRIDE + c0 + l];
        acc = __builtin_amdgcn_wmma_f32_16x16x4_f32(
            /*neg_a=*/false, a, /*neg_b=*/false, b,
            /*c_mod=*/(short)0, acc, /*reuse_a=*/false, /*reuse_b=*/false);
    }

    // D layout: VGPR i -> row (half*8 + i), col = lane&15 within the tile.
#pragma unroll
    for (int i = 0; i < 8; ++i) {
        const int m = half * 8 + i;
        S[(size_t)(r0 + m) * DD + c0 + l] = acc[i];
    }
}

// ---------------------------------------------------------------------------
// out = 0.9 * init   (pre-seeds the residual blend; float4, coalesced)
// ---------------------------------------------------------------------------
__global__ __launch_bounds__(256) void blend_init(
    const float4* __restrict__ init, float4* __restrict__ out, int n4)
{
    const int i = blockIdx.x * blockDim.x + threadIdx.x;
    if (i < n4) {
        float4 v = init[i];
        v.x *= ONE_M_ALPHA; v.y *= ONE_M_ALPHA;
        v.z *= ONE_M_ALPHA; v.w *= ONE_M_ALPHA;
        out[i] = v;
    }
}

// ---------------------------------------------------------------------------
// out[row[e]] += 0.1 * val[e] * support[col[e]]   (one wave per edge;
// float4 gather per lane + 4 global f32 atomics)
// ---------------------------------------------------------------------------
__global__ __launch_bounds__(256) void edge_scatter(
    const float* __restrict__ S, const int* __restrict__ erow,
    const int* __restrict__ ecol, const float* __restrict__ eval,
    float* __restrict__ out, int nEdges)
{
    const int e = blockIdx.x * 8 + (threadIdx.x >> 5);
    if (e >= nEdges) return;
    const int lane = threadIdx.x & 31;
    const int row  = erow[e];
    const int col  = ecol[e];
    const float s  = ALPHA_F * eval[e];

    const float4 v = *(const float4*)(S + (size_t)col * DD + lane * 4);
    float* dst = out + (size_t)row * DD + lane * 4;
    atomicAdd(dst + 0, s * v.x);
    atomicAdd(dst + 1, s * v.y);
    atomicAdd(dst + 2, s * v.z);
    atomicAdd(dst + 3, s * v.w);
}

// ---------------------------------------------------------------------------
// Final: gather+ReLU code embedding, RNN cell (tanh), row-wise L2 normalize.
// One block (128 threads) per batch row b; thread j = output feature.
// ---------------------------------------------------------------------------
__global__ __launch_bounds__(128) void rnn_final(
    const float* __restrict__ x2, const float* __restrict__ pat_dyn,
    const float* __restrict__ timed, const float* __restrict__ feats,
    const float* __restrict__ W_ih, const float* __restrict__ b_ih,
    const float* __restrict__ W_hh, const float* __restrict__ b_hh,
    const int* __restrict__ codeid, const int* __restrict__ patientid,
    float* __restrict__ out)
{
    __shared__ float s_pat[DD], s_ce[DD], s_feat[FEATSZ], red[DD];

    const int b = blockIdx.x;
    const int j = threadIdx.x;
    const int pid = patientid[0];
    const int cid = codeid[b];

    s_pat[j]  = pat_dyn[(size_t)pid * DD + j];
    s_ce[j]   = fmaxf(x2[(size_t)cid * DD + j], 0.0f);   // ReLU of layer-2 out
    s_feat[j] = feats[(size_t)b * FEATSZ + j];
    __syncthreads();

    float acc = b_ih[j] + b_hh[j];
    const float* wih = W_ih + (size_t)j * IN_RNN;
#pragma unroll 4
    for (int k = 0; k < DD; ++k)      acc += s_pat[k]  * wih[k];
    acc += timed[b] * wih[DD];
#pragma unroll 4
    for (int k = 0; k < FEATSZ; ++k)  acc += s_feat[k] * wih[DD + 1 + k];
    const float* whh = W_hh + (size_t)j * DD;
#pragma unroll 4
    for (int k = 0; k < DD; ++k)      acc += s_ce[k]   * whh[k];

    const float h = tanhf(acc);

    red[j] = h * h;
    __syncthreads();
    for (int s = 64; s > 0; s >>= 1) {
        if (j < s) red[j] += red[j + s];
        __syncthreads();
    }
    const float nrm = fmaxf(sqrtf(red[0]), EPS_F);
    out[(size_t)b * DD + j] = h / nrm;
}

// ---------------------------------------------------------------------------
// Host launcher
// ---------------------------------------------------------------------------
extern "C" void kernel_launch(void* const* d_in, const int* in_sizes, int n_in,
                              void* d_out, int out_size, void* d_ws, size_t ws_size,
                              hipStream_t stream)
{
    (void)in_sizes; (void)n_in; (void)out_size; (void)ws_size;

    const float* code_dyn  = (const float*)d_in[0];
    const float* init_dyn  = (const float*)d_in[1];
    const float* pat_dyn   = (const float*)d_in[2];
    const float* timed     = (const float*)d_in[3];
    const float* feats     = (const float*)d_in[4];
    const float* eval      = (const float*)d_in[5];
    const float* W1        = (const float*)d_in[6];
    const float* W2        = (const float*)d_in[7];
    const float* W_ih      = (const float*)d_in[8];
    const float* b_ih      = (const float*)d_in[9];
    const float* W_hh      = (const float*)d_in[10];
    const float* b_hh      = (const float*)d_in[11];
    const int*   erow      = (const int*)d_in[12];
    const int*   ecol      = (const int*)d_in[13];
    const int*   codeid    = (const int*)d_in[14];
    const int*   patientid = (const int*)d_in[15];
    float*       out       = (float*)d_out;

    // Workspace: two N*D fp32 buffers (ping-pong), 61.4 MB total.
    float* support = (float*)d_ws;                       // GEMM output
    float* xbuf    = support + (size_t)NN * DD;          // blended/aggregated x

    const int n4        = (NN * DD) / 4;                 // 1,920,000 float4
    const int initBlks  = (n4 + 255) / 256;              // 7500
    const int gemmBlks  = NN / 16;                       // 3750
    const int scatBlks  = (EE + 7) / 8;                  // 75000

    // ---- layer 1: x1 = relu(0.9*init + 0.1*A@(x @ W1)) ----
    blend_init   <<<initBlks, 256, 0, stream>>>((const float4*)init_dyn, (float4*)xbuf, n4);
    gcn_gemm_wmma<<<gemmBlks, 256, 0, stream>>>(code_dyn, W1, support, /*relu=*/0);
    edge_scatter <<<scatBlks, 256, 0, stream>>>(support, erow, ecol, eval, xbuf, EE);
    // (ReLU of layer 1 fused into layer-2 GEMM A-load)

    // ---- layer 2: x2 = relu(0.9*init + 0.1*A@(relu(x1) @ W2)) ----
    gcn_gemm_wmma<<<gemmBlks, 256, 0, stream>>>(xbuf, W2, support, /*relu=*/1);
    blend_init   <<<initBlks, 256, 0, stream>>>((const float4*)init_dyn, (float4*)xbuf, n4);
    edge_scatter <<<scatBlks, 256, 0, stream>>>(support, erow, ecol, eval, xbuf, EE);
    // (ReLU of layer 2 fused into rnn_final gather)

    // ---- RNN cell + L2 normalize ----
    rnn_final<<<BB, DD, 0, stream>>>(xbuf, pat_dyn, timed, feats,
                                     W_ih, b_ih, W_hh, b_hh,
                                     codeid, patientid, out);
}